// RoPETransform_70368744178158
// MI455X (gfx1250) — compile-verified
//
#include <hip/hip_runtime.h>

typedef float v2f __attribute__((ext_vector_type(2)));
typedef float v4f __attribute__((ext_vector_type(4)));
typedef float v8f __attribute__((ext_vector_type(8)));

#define NFREQ           16
#define SITES_PER_TILE  16
#define NTHREADS        256
#define TILE_FLOATS     (SITES_PER_TILE * 256)   // 4096 floats = 16 KB
#define MAX_BLOCKS      512

// Persistent workgroups: setup (Cayley Q_f -> M_f = Q_f (x) Q_f) once, then a
// grid-stride loop over 16-site tiles with double-buffered LDS staging.
// Per (f, 16 sites): D[16x16] = M_f[16x16] * Rcols[16x16] via 4 chained
// V_WMMA_F32_16X16X4_F32  (vec_row(Q R Q^T) = (Q (x) Q) vec_row(R)).
__global__ __launch_bounds__(NTHREADS)
void rope_cayley_wmma_kernel(const float* __restrict__ skew,
                             const float* __restrict__ rope,
                             float* __restrict__ out,
                             int total_sites)
{
    // [ M: 16x256 | buf0: 4096 | buf1: 4096 ] = 48 KB
    __shared__ float smem[NFREQ * 256 + 2 * TILE_FLOATS];
    float* ldsM = smem;
    float* bufs = smem + NFREQ * 256;
    float* Qbuf = bufs;  // 256 floats, reused before the tile loop starts

    const int tid = threadIdx.x;

    // ---- Cayley: thread f computes Q_f = (I - S)(I + S)^-1, writes 16 floats ----
    if (tid < NFREQ) {
        const int f = tid;
        const float* sp = skew + f * 6;
        float S[4][4];
#pragma unroll
        for (int i = 0; i < 4; ++i)
#pragma unroll
            for (int j = 0; j < 4; ++j) S[i][j] = 0.0f;
        // triu_indices(4,1): (0,1)(0,2)(0,3)(1,2)(1,3)(2,3)
        S[0][1] =  sp[0]; S[0][2] =  sp[1]; S[0][3] =  sp[2];
        S[1][2] =  sp[3]; S[1][3] =  sp[4]; S[2][3] =  sp[5];
        S[1][0] = -sp[0]; S[2][0] = -sp[1]; S[3][0] = -sp[2];
        S[2][1] = -sp[3]; S[3][1] = -sp[4]; S[3][2] = -sp[5];

        float A[4][4], V[4][4];
#pragma unroll
        for (int i = 0; i < 4; ++i)
#pragma unroll
            for (int j = 0; j < 4; ++j) {
                A[i][j] = ((i == j) ? 1.0f : 0.0f) + S[i][j];
                V[i][j] = (i == j) ? 1.0f : 0.0f;
            }
        for (int col = 0; col < 4; ++col) {
            int p = col;
            float best = fabsf(A[col][col]);
            for (int r = col + 1; r < 4; ++r) {
                float v = fabsf(A[r][col]);
                if (v > best) { best = v; p = r; }
            }
            if (p != col) {
                for (int j = 0; j < 4; ++j) {
                    float t = A[col][j]; A[col][j] = A[p][j]; A[p][j] = t;
                    t = V[col][j]; V[col][j] = V[p][j]; V[p][j] = t;
                }
            }
            const float inv = 1.0f / A[col][col];
            for (int j = 0; j < 4; ++j) { A[col][j] *= inv; V[col][j] *= inv; }
            for (int r = 0; r < 4; ++r) {
                if (r == col) continue;
                const float fac = A[r][col];
                for (int j = 0; j < 4; ++j) {
                    A[r][j] -= fac * A[col][j];
                    V[r][j] -= fac * V[col][j];
                }
            }
        }
#pragma unroll
        for (int i = 0; i < 4; ++i) {
            v4f q;
#pragma unroll
            for (int j = 0; j < 4; ++j) {
                float acc = V[i][j];
#pragma unroll
                for (int k = 0; k < 4; ++k) acc -= S[i][k] * V[k][j];
                q[j] = acc;
            }
            *(v4f*)(Qbuf + f * 16 + i * 4) = q;
        }
    }
    __syncthreads();

    // ---- Kronecker M_f = Q_f (x) Q_f, fully parallel: 1 row per thread ----
    {
        const int f = tid >> 4, m = tid & 15;
        const int i = m >> 2, k = m & 3;
        const v4f Qi = *(const v4f*)(Qbuf + f * 16 + i * 4);
        const v4f Qk = *(const v4f*)(Qbuf + f * 16 + k * 4);
        float* dst = ldsM + f * 256 + m * 16;
#pragma unroll
        for (int j = 0; j < 4; ++j) {
            v4f w = { Qi[j] * Qk[0], Qi[j] * Qk[1], Qi[j] * Qk[2], Qi[j] * Qk[3] };
            *(v4f*)(dst + j * 4) = w;
        }
    }
    __syncthreads();   // M ready; Qbuf region may now be reused as buf0

    const long total_elems = (long)total_sites * 256;
    const int  tiles  = (total_sites + SITES_PER_TILE - 1) / SITES_PER_TILE;
    const int  stride = gridDim.x;

    const int lane = tid & 31;
    const int wave = tid >> 5;
    const int row  = lane & 15;            // A-row / site / D-column
    const int cofs = (lane < 16) ? 0 : 2;  // K-pair: lanes 0-15 -> K0,1 ; 16-31 -> K2,3

    // ---- software pipeline: regs hold tile t while tile t+stride loads ----
    long t = blockIdx.x;
    v4f r0, r1, r2, r3;
    if (t < tiles) {
        long e = t * TILE_FLOATS + (long)tid * 16;
        if (e + 16 > total_elems) e = 0;  // clamp (sites never stored)
        const v4f* g = (const v4f*)(rope + e);
        r0 = g[0]; r1 = g[1]; r2 = g[2]; r3 = g[3];
    }

    int p = 0;
    while (t < tiles) {
        float* buf = bufs + p * TILE_FLOATS;
        {
            v4f* l = (v4f*)(buf + tid * 16);
            l[0] = r0; l[1] = r1; l[2] = r2; l[3] = r3;
        }
        __syncthreads();

        // issue next tile's loads now; their wait lands after the WMMA block
        const long tn = t + stride;
        if (tn < tiles) {
            long e = tn * TILE_FLOATS + (long)tid * 16;
            if (e + 16 > total_elems) e = 0;
            const v4f* g = (const v4f*)(rope + e);
            r0 = g[0]; r1 = g[1]; r2 = g[2]; r3 = g[3];
            const long e2 = e + (long)stride * TILE_FLOATS;  // one more tile ahead
            if (e2 + 16 <= total_elems) __builtin_prefetch(rope + e2, 0, 1);
        }

        const long site0 = t * SITES_PER_TILE;
#pragma unroll
        for (int ff = 0; ff < 2; ++ff) {
            const int f = wave * 2 + ff;
            v8f acc = {};
#pragma unroll
            for (int k = 0; k < 4; ++k) {
                // A fragment: M_f rows striped over lanes, K-pair per VGPR pair
                v2f a = *(const v2f*)(ldsM + f * 256 + row * 16 + 4 * k + cofs);
                // B fragment: column 'row' = site 'row', elements 4k+cofs, +1
                v2f b = *(const v2f*)(buf  + row * 256 + f * 16 + 4 * k + cofs);
                acc = __builtin_amdgcn_wmma_f32_16x16x4_f32(
                    false, a, false, b, (short)0, acc, false, false);
            }
            // D: lane<16 holds rows 0..7 of column 'row'; lane>=16 rows 8..15
            if (site0 + row < (long)total_sites) {
                float* dst = out + (site0 + row) * 256 + f * 16 + ((lane < 16) ? 0 : 8);
                v4f lo = { acc[0], acc[1], acc[2], acc[3] };
                v4f hi = { acc[4], acc[5], acc[6], acc[7] };
                *(v4f*)dst = lo;
                *(v4f*)(dst + 4) = hi;
            }
        }
        t = tn;
        p ^= 1;
    }
}

extern "C" void kernel_launch(void* const* d_in, const int* in_sizes, int n_in,
                              void* d_out, int out_size, void* d_ws, size_t ws_size,
                              hipStream_t stream) {
    (void)n_in; (void)out_size; (void)d_ws; (void)ws_size;
    const float* skew = (const float*)d_in[0];   // [16, 6] f32
    const float* rope = (const float*)d_in[1];   // [4, 8192, 16, 4, 4] f32
    float* out = (float*)d_out;

    const int total_sites = in_sizes[1] / 256;   // (b,s) pairs: 32768
    const int tiles = (total_sites + SITES_PER_TILE - 1) / SITES_PER_TILE;
    const int blocks = (tiles < MAX_BLOCKS) ? tiles : MAX_BLOCKS;

    rope_cayley_wmma_kernel<<<blocks, NTHREADS, 0, stream>>>(skew, rope, out,
                                                             total_sites);
}